// BipartiteSAGE_48344151884188
// MI455X (gfx1250) — compile-verified
//
#include <hip/hip_runtime.h>

// Problem constants (match reference)
#define N_POL   100000
#define N_TICK  20000
#define NN      (N_POL + N_TICK)   // 120000 nodes, divisible by 16
#define EDGES   1000000
#define POL_F   7
#define EMB     128
#define HID     256
#define OUT_D   128

typedef __attribute__((ext_vector_type(2))) float v2f;
typedef __attribute__((ext_vector_type(8))) float v8f;

// ---------------------------------------------------------------------------
// Stage 0: x[0:N_POL]  = relu(pol_features @ W_proj + b_proj) + state_emb[sid]
//          x[N_POL:NN] = emb_tick
// One thread per (node, dim). 7-term dot per thread; coalesced on dim.
// ---------------------------------------------------------------------------
__global__ void build_x_kernel(const float* __restrict__ pf,
                               const int*   __restrict__ sid,
                               const float* __restrict__ Wp,
                               const float* __restrict__ bp,
                               const float* __restrict__ semb,
                               const float* __restrict__ tick,
                               float* __restrict__ x) {
  long idx = (long)blockIdx.x * blockDim.x + threadIdx.x;
  if (idx >= (long)NN * EMB) return;
  int i = (int)(idx / EMB);
  int j = (int)(idx % EMB);
  if (i < N_POL) {
    float acc = bp[j];
#pragma unroll
    for (int f = 0; f < POL_F; ++f)
      acc += pf[(long)i * POL_F + f] * Wp[f * EMB + j];
    acc = acc > 0.f ? acc : 0.f;
    x[idx] = acc + semb[(long)sid[i] * EMB + j];
  } else {
    x[idx] = tick[(long)(i - N_POL) * EMB + j];
  }
}

__global__ void zero_kernel(float4* __restrict__ p, long n4) {
  long i = (long)blockIdx.x * blockDim.x + threadIdx.x;
  if (i < n4) p[i] = make_float4(0.f, 0.f, 0.f, 0.f);
}

// ---------------------------------------------------------------------------
// Edge scatter: agg[dst] += w * x[src]   (segment_sum)
// D/4 threads per edge, each owning 4 contiguous dims -> float4 read +
// 4 hardware fp32 atomics. x/agg are L2-resident (192MB L2), so the atomics
// resolve in L2, which is the bandwidth-optimal path on MI455X.
// ---------------------------------------------------------------------------
template <int D>
__global__ void scatter_kernel(const int*   __restrict__ ei,   // [2,E]
                               const float* __restrict__ ew,   // [E]
                               const float* __restrict__ x,    // [NN,D]
                               float* __restrict__ agg) {      // [NN,D]
  const int TPE = D / 4;
  long t = (long)blockIdx.x * blockDim.x + threadIdx.x;
  long e = t / TPE;
  if (e >= EDGES) return;
  int lane4 = (int)(t % TPE) * 4;
  int src = ei[e];
  int dst = ei[EDGES + e];
  float w = ew[e];
  const float4 v = *(const float4*)(x + (long)src * D + lane4);
  float* a = agg + (long)dst * D + lane4;
  unsafeAtomicAdd(a + 0, w * v.x);
  unsafeAtomicAdd(a + 1, w * v.y);
  unsafeAtomicAdd(a + 2, w * v.z);
  unsafeAtomicAdd(a + 3, w * v.w);
}

// ---------------------------------------------------------------------------
// Fused dual GEMM:  C = act( A1 @ W1 + A2 @ W2 + bias )
// One wave per 16x16 output tile, V_WMMA_F32_16X16X4_F32 over K in steps
// of 4 for both terms (fp32 in/out: bit-compatible with the fp32 reference,
// and the GEMMs are a tiny fraction of runtime vs the edge scatter anyway).
//
// ISA fragment layouts (wave32, CDNA5 7.12.2):
//   A 16x4 : lane l -> row m0+(l&15), VGPR{0,1} = K = k+2*(l>>4) , +1
//   B 4x16 : lane l -> col n0+(l&15), VGPR{0,1} = K = k+2*(l>>4) , +1
//   C 16x16: lane l -> col n0+(l&15), VGPR v    = row m0 + v + 8*(l>>4)
// Grids are exact (tiles % 8 waves == 0) so EXEC is all-ones for every WMMA.
// ---------------------------------------------------------------------------
template <int K, int NOUT, bool RELU>
__global__ __launch_bounds__(256)
void gemm_dual_kernel(const float* __restrict__ A1, const float* __restrict__ W1,
                      const float* __restrict__ A2, const float* __restrict__ W2,
                      const float* __restrict__ bias,
                      float* __restrict__ C) {
  const int wave = threadIdx.x >> 5;
  const int lane = threadIdx.x & 31;
  const int half = lane >> 4;     // 0: K+{0,1}, 1: K+{2,3}
  const int mrem = lane & 15;

  const int nTilesN = NOUT >> 4;
  const int tile = blockIdx.x * 8 + wave;
  const int m0 = (tile / nTilesN) * 16;
  const int n0 = (tile % nTilesN) * 16;
  const int n  = n0 + mrem;

  v8f c = {};

  const float* a1row = A1 + (long)(m0 + mrem) * K;
#pragma unroll
  for (int k = 0; k < K; k += 4) {
    const int kb = k + half * 2;
    v2f a = *(const v2f*)(a1row + kb);        // 8B-aligned (kb even)
    v2f b;
    b.x = W1[(long)(kb)     * NOUT + n];
    b.y = W1[(long)(kb + 1) * NOUT + n];
    c = __builtin_amdgcn_wmma_f32_16x16x4_f32(false, a, false, b,
                                              (short)0, c, false, false);
  }

  const float* a2row = A2 + (long)(m0 + mrem) * K;
#pragma unroll
  for (int k = 0; k < K; k += 4) {
    const int kb = k + half * 2;
    v2f a = *(const v2f*)(a2row + kb);
    v2f b;
    b.x = W2[(long)(kb)     * NOUT + n];
    b.y = W2[(long)(kb + 1) * NOUT + n];
    c = __builtin_amdgcn_wmma_f32_16x16x4_f32(false, a, false, b,
                                              (short)0, c, false, false);
  }

  const float bn = bias[n];
#pragma unroll
  for (int v = 0; v < 8; ++v) {
    float val = c[v] + bn;
    if (RELU) val = val > 0.f ? val : 0.f;
    C[(long)(m0 + v + 8 * half) * NOUT + n] = val;
  }
}

// ---------------------------------------------------------------------------
extern "C" void kernel_launch(void* const* d_in, const int* in_sizes, int n_in,
                              void* d_out, int out_size, void* d_ws, size_t ws_size,
                              hipStream_t stream) {
  const float* pol_features = (const float*)d_in[0];
  const int*   state_ids    = (const int*)  d_in[1];
  const int*   edge_index   = (const int*)  d_in[2];
  const float* edge_weight  = (const float*)d_in[3];
  const float* W_proj       = (const float*)d_in[4];
  const float* b_proj       = (const float*)d_in[5];
  const float* state_emb    = (const float*)d_in[6];
  const float* emb_tick     = (const float*)d_in[7];
  const float* W1_rel       = (const float*)d_in[8];
  const float* b1_rel       = (const float*)d_in[9];
  const float* W1_root      = (const float*)d_in[10];
  const float* W2_rel       = (const float*)d_in[11];
  const float* b2_rel       = (const float*)d_in[12];
  const float* W2_root      = (const float*)d_in[13];
  float* out = (float*)d_out;

  // Workspace layout (floats): x[NN*128] | h[NN*256] | agg[NN*256]
  float* x   = (float*)d_ws;
  float* h   = x + (long)NN * EMB;
  float* agg = h + (long)NN * HID;

  // Stage 0: node features
  build_x_kernel<<<(NN * EMB) / 256, 256, 0, stream>>>(
      pol_features, state_ids, W_proj, b_proj, state_emb, emb_tick, x);

  // Layer 1: agg1 = segment_sum(w * x[src] -> dst)  (D=128)
  zero_kernel<<<((long)NN * EMB / 4 + 255) / 256, 256, 0, stream>>>(
      (float4*)agg, (long)NN * EMB / 4);
  scatter_kernel<EMB><<<(EDGES * (EMB / 4) + 255) / 256, 256, 0, stream>>>(
      edge_index, edge_weight, x, agg);
  // h = relu(agg @ W1_rel + x @ W1_root + b1)   [120000 x 256]
  gemm_dual_kernel<EMB, HID, true>
      <<<(NN / 16) * (HID / 16) / 8, 256, 0, stream>>>(
          agg, W1_rel, x, W1_root, b1_rel, h);

  // Layer 2: agg2 = segment_sum(w * h[src] -> dst)  (D=256)
  zero_kernel<<<((long)NN * HID / 4 + 255) / 256, 256, 0, stream>>>(
      (float4*)agg, (long)NN * HID / 4);
  scatter_kernel<HID><<<(EDGES * (HID / 4) + 255) / 256, 256, 0, stream>>>(
      edge_index, edge_weight, h, agg);
  // out = agg @ W2_rel + h @ W2_root + b2   [120000 x 128]
  gemm_dual_kernel<HID, OUT_D, false>
      <<<(NN / 16) * (OUT_D / 16) / 8, 256, 0, stream>>>(
          agg, W2_rel, h, W2_root, b2_rel, out);
}